// GraphAttentionLayer_2894807957836
// MI455X (gfx1250) — compile-verified
//
#include <hip/hip_runtime.h>
#include <hip/hip_bf16.h>
#include <stddef.h>
#include <stdint.h>

// GraphAttention: B=4, N=512, HIDDEN=128, HEADS=8, HEAD_DIM=16
// Memory-bound on edge_features (536MB -> ~23us at 23.3 TB/s).
// GEMMs via V_WMMA_F32_16X16X4_F32; edge tile staged with
// GLOBAL_LOAD_ASYNC_TO_LDS_B128 (ASYNCcnt) on the hot path.

typedef float v2f __attribute__((ext_vector_type(2)));
typedef float v8f __attribute__((ext_vector_type(8)));
typedef int   v4i __attribute__((ext_vector_type(4)));

static __device__ __forceinline__ v8f wmma4(v2f a, v2f b, v8f c) {
  // D(16x16,f32) = A(16x4,f32) * B(4x16,f32) + C
  return __builtin_amdgcn_wmma_f32_16x16x4_f32(false, a, false, b, (short)0, c, false, false);
}

#ifndef __has_builtin
#define __has_builtin(x) 0
#endif

// ---- CDNA5 async global->LDS copy (B128) + ASYNCcnt wait -------------------
static __device__ __forceinline__ void async_copy_b128(const float* src, float* lds_dst) {
#if __has_builtin(__builtin_amdgcn_global_load_async_to_lds_b128)
  __attribute__((address_space(1))) void* g0 =
      (__attribute__((address_space(1))) void*)(void*)const_cast<float*>(src);
  __attribute__((address_space(3))) void* l0 =
      (__attribute__((address_space(3))) void*)(void*)lds_dst;
  __builtin_amdgcn_global_load_async_to_lds_b128(
      (__attribute__((address_space(1))) v4i*)g0,
      (__attribute__((address_space(3))) v4i*)l0, 0, 0);
#else
  const uint64_t ga = (uint64_t)(uintptr_t)src;
  const uint32_t la =
      (uint32_t)(uintptr_t)(__attribute__((address_space(3))) float*)lds_dst;
  asm volatile("global_load_async_to_lds_b128 %0, %1, off"
               :: "v"(la), "v"(ga) : "memory");
#endif
}

static __device__ __forceinline__ void wait_asynccnt0() {
#if __has_builtin(__builtin_amdgcn_s_wait_asynccnt)
  __builtin_amdgcn_s_wait_asynccnt(0);
#else
  asm volatile("s_wait_asynccnt 0x0" ::: "memory");
#endif
}

#define NB   4
#define NN   512
#define HID  128
#define NH   8
#define HD   16

// ---------------------------------------------------------------------------
// Kernel 1: QKV projections. grid=(128 row-tiles, 3), block=256 (8 waves).
// Wave w computes output column tile w (16 cols = one head).
// Q,K stored [b][h][n][d]; V stored transposed [b][h][d][n].
// ---------------------------------------------------------------------------
__global__ void qkv_proj(const float* __restrict__ x,
                         const float* __restrict__ Wq, const float* __restrict__ bq,
                         const float* __restrict__ Wk, const float* __restrict__ bk,
                         const float* __restrict__ Wv, const float* __restrict__ bv,
                         float* __restrict__ Q, float* __restrict__ Kq,
                         float* __restrict__ Vt) {
  const int mode = blockIdx.y;               // 0=Q 1=K 2=V
  const float* W    = (mode == 0) ? Wq : (mode == 1) ? Wk : Wv;
  const float* bias = (mode == 0) ? bq : (mode == 1) ? bk : bv;
  const int rt   = blockIdx.x;               // 16-row tile over B*N
  const int wv   = threadIdx.x >> 5;
  const int lane = threadIdx.x & 31;
  const int l15  = lane & 15;
  const int lh   = lane >> 4;

  const int arow = rt * 16 + l15;            // A-fragment row (node)
  const int jcol = wv * 16 + l15;            // B/D column (feature)

  v8f c = {};
#pragma unroll 4
  for (int kc = 0; kc < 32; ++kc) {
    const int k = kc * 4 + lh * 2;
    v2f a; a.x = x[arow * HID + k];     a.y = x[arow * HID + k + 1];
    v2f b; b.x = W[k * HID + jcol];     b.y = W[(k + 1) * HID + jcol];
    c = wmma4(a, b, c);
  }
  const float bcol = bias[jcol];
  const int h = wv, d = l15;
#pragma unroll
  for (int r = 0; r < 8; ++r) {
    const int M = r + lh * 8;
    const int g = rt * 16 + M;               // global row in [0, B*N)
    const int bb = g >> 9, nn = g & 511;
    const float val = c[r] + bcol;
    if (mode == 0)      Q [((bb * NH + h) * NN + nn) * HD + d] = val;
    else if (mode == 1) Kq[((bb * NH + h) * NN + nn) * HD + d] = val;
    else                Vt[((bb * NH + h) * HD + d) * NN + nn] = val;
  }
}

// ---------------------------------------------------------------------------
// Kernel 2: scores[b,h,i,j] = QK^T*scale + (E@We + be) masked.
// grid=(32 j-tiles, 32 i-tiles, 4), block=256 (8 waves).
// Edge tile (256 edges x 128 feats) staged via async-to-LDS in 4 chunks of 32.
// ---------------------------------------------------------------------------
#define EPAD 36   // 32 + 4 pad: fragment reads hit 32 distinct LDS banks

__global__ void scores_kernel(const float* __restrict__ E,
                              const unsigned char* __restrict__ mask,
                              const float* __restrict__ We,
                              const float* __restrict__ be,
                              const float* __restrict__ Q,
                              const float* __restrict__ Kq,
                              float* __restrict__ S) {
  __shared__ float lds_e[256 * EPAD];        // 36 KB
  __shared__ float lds_bias[256 * 9];        // 9 KB, pad 9 -> conflict-free read

  const int jt = blockIdx.x, it = blockIdx.y, b = blockIdx.z;
  const int i0 = it * 16, j0 = jt * 16;
  const int tid  = threadIdx.x;
  const int wv   = tid >> 5;
  const int lane = tid & 31;
  const int l15  = lane & 15;
  const int lh   = lane >> 4;

  // ---- edge-bias GEMM: (256 x 128) @ (128 x 16), wave wv owns edges [32wv,32wv+32)
  v8f c0 = {}, c1 = {};
  for (int cc = 0; cc < 4; ++cc) {
    __syncthreads();                         // LDS buffer free for reuse
    // async copy: 256 rows x 32 floats (8 x b128 per thread, 4x128B lines/instr)
#pragma unroll
    for (int t2 = 0; t2 < 8; ++t2) {
      const int idx = tid + t2 * 256;        // 0..2047
      const int e = idx >> 3, q4 = idx & 7;
      const int i = i0 + (e >> 4), j = j0 + (e & 15);
      async_copy_b128(&E[(((size_t)b * NN + i) * NN + j) * HID + cc * 32 + q4 * 4],
                      &lds_e[e * EPAD + q4 * 4]);
    }
    wait_asynccnt0();                        // this wave's chunk is in LDS
    __syncthreads();                         // ... and every other wave's too
#pragma unroll
    for (int kc = 0; kc < 8; ++kc) {
      const int k  = kc * 4 + lh * 2;        // k within chunk
      const int gk = cc * 32 + k;            // global feature index
      v2f bf;
      bf.x = (l15 < NH) ? We[gk * NH + l15] : 0.0f;
      bf.y = (l15 < NH) ? We[(gk + 1) * NH + l15] : 0.0f;
      const int e0 = wv * 32 + l15;
      v2f a0; a0.x = lds_e[e0 * EPAD + k]; a0.y = lds_e[e0 * EPAD + k + 1];
      c0 = wmma4(a0, bf, c0);
      const int e1 = e0 + 16;
      v2f a1; a1.x = lds_e[e1 * EPAD + k]; a1.y = lds_e[e1 * EPAD + k + 1];
      c1 = wmma4(a1, bf, c1);
    }
  }
  // scatter bias (+be) into LDS as bias[e][h]
  if (l15 < NH) {
    const float beh = be[l15];
#pragma unroll
    for (int r = 0; r < 8; ++r) {
      const int M = r + lh * 8;
      lds_bias[(wv * 32 + M) * 9 + l15]      = c0[r] + beh;
      lds_bias[(wv * 32 + 16 + M) * 9 + l15] = c1[r] + beh;
    }
  }
  __syncthreads();

  // ---- Q.K^T for head h = wv (16x16 tile, K=16 -> 4 WMMAs)
  const int h = wv;
  const float* Qh = &Q [((size_t)b * NH + h) * NN * HD];
  const float* Kh = &Kq[((size_t)b * NH + h) * NN * HD];
  v8f s = {};
#pragma unroll
  for (int kc = 0; kc < 4; ++kc) {
    const int k = kc * 4 + lh * 2;
    v2f a;  a.x  = Qh[(i0 + l15) * HD + k]; a.y  = Qh[(i0 + l15) * HD + k + 1];
    v2f bf; bf.x = Kh[(j0 + l15) * HD + k]; bf.y = Kh[(j0 + l15) * HD + k + 1]; // B = K^T
    s = wmma4(a, bf, s);
  }
  const float scale = 0.25f;                 // 1/sqrt(HEAD_DIM=16)
#pragma unroll
  for (int r = 0; r < 8; ++r) {
    const int M = r + lh * 8;
    const int i = i0 + M, j = j0 + l15;
    float val = s[r] * scale + lds_bias[(M * 16 + l15) * 9 + h];
    const unsigned char mk = mask[((size_t)b * NN + i) * NN + j];
    val = mk ? val : -__builtin_inff();
    S[(((size_t)b * NH + h) * NN + i) * NN + j] = val;
  }
}

// ---------------------------------------------------------------------------
// Kernel 3: softmax over keys + (w @ V). grid=(32 i-tiles, 8 heads, 4), block=256.
// ---------------------------------------------------------------------------
#define SPAD 516  // 512 + 4 pad -> WMMA A-fragment reads conflict-free

__global__ void softmax_av(const float* __restrict__ S,
                           const float* __restrict__ Vt,
                           float* __restrict__ attn) {
  __shared__ float w_lds[16 * SPAD];         // ~32 KB
  __shared__ float pads[8 * 256];            // 8 KB partial D tiles
  __shared__ float rowmax[16], rowsum[16];

  const int it = blockIdx.x, h = blockIdx.y, b = blockIdx.z;
  const int i0 = it * 16;
  const int tid = threadIdx.x;
  const float* Srow = &S[(((size_t)b * NH + h) * NN + i0) * NN];

  for (int idx = tid; idx < 16 * NN; idx += 256) {
    const int r = idx >> 9, cc = idx & 511;
    w_lds[r * SPAD + cc] = Srow[r * NN + cc];
  }
  __syncthreads();

  const int r = tid >> 4, sub = tid & 15;    // 16 threads per row
  const float NEG_INF = -__builtin_inff();
  float m = NEG_INF;
  for (int cc = sub; cc < NN; cc += 16) m = fmaxf(m, w_lds[r * SPAD + cc]);
#pragma unroll
  for (int off = 8; off > 0; off >>= 1) m = fmaxf(m, __shfl_xor(m, off, 16));
  if (sub == 0) rowmax[r] = m;
  __syncthreads();

  const float mr = rowmax[r];
  float ssum = 0.0f;
  for (int cc = sub; cc < NN; cc += 16) {
    const float p = (mr == NEG_INF) ? 0.0f : __expf(w_lds[r * SPAD + cc] - mr);
    w_lds[r * SPAD + cc] = p;
    ssum += p;
  }
#pragma unroll
  for (int off = 8; off > 0; off >>= 1) ssum += __shfl_xor(ssum, off, 16);
  if (sub == 0) rowsum[r] = ssum;
  __syncthreads();

  const float inv = (rowsum[r] > 0.0f) ? (1.0f / rowsum[r]) : 0.0f;
  for (int cc = sub; cc < NN; cc += 16) w_lds[r * SPAD + cc] *= inv;
  __syncthreads();

  // w(16x512) @ V(512x16): wave wv owns K-slice [64wv, 64wv+64)
  const int wv = tid >> 5, lane = tid & 31, l15 = lane & 15, lh = lane >> 4;
  const float* Vh = &Vt[((size_t)b * NH + h) * HD * NN]; // [d][key]
  v8f c = {};
#pragma unroll 4
  for (int kc = 0; kc < 16; ++kc) {
    const int k = wv * 64 + kc * 4 + lh * 2;
    v2f a;  a.x  = w_lds[l15 * SPAD + k]; a.y  = w_lds[l15 * SPAD + k + 1];
    v2f bf; bf.x = Vh[l15 * NN + k];      bf.y = Vh[l15 * NN + k + 1]; // B[k][n]=Vt[n][k]
    c = wmma4(a, bf, c);
  }
#pragma unroll
  for (int rr = 0; rr < 8; ++rr) {
    const int M = rr + lh * 8;
    pads[wv * 256 + M * 16 + l15] = c[rr];
  }
  __syncthreads();

  const int orow = tid >> 4, ocol = tid & 15;
  float acc = 0.0f;
#pragma unroll
  for (int w2 = 0; w2 < 8; ++w2) acc += pads[w2 * 256 + orow * 16 + ocol];
  attn[((size_t)b * NN + i0 + orow) * HID + h * HD + ocol] = acc;
}

// ---------------------------------------------------------------------------
// Kernel 4: output projection (2048x128)@(128x128)+bo -> d_out (fp32).
// ---------------------------------------------------------------------------
__global__ void out_proj(const float* __restrict__ attn,
                         const float* __restrict__ Wo,
                         const float* __restrict__ bo,
                         float* __restrict__ out) {
  const int rt   = blockIdx.x;
  const int wv   = threadIdx.x >> 5;
  const int lane = threadIdx.x & 31;
  const int l15  = lane & 15;
  const int lh   = lane >> 4;
  const int arow = rt * 16 + l15;
  const int jcol = wv * 16 + l15;

  v8f c = {};
#pragma unroll 4
  for (int kc = 0; kc < 32; ++kc) {
    const int k = kc * 4 + lh * 2;
    v2f a; a.x = attn[arow * HID + k]; a.y = attn[arow * HID + k + 1];
    v2f b; b.x = Wo[k * HID + jcol];   b.y = Wo[(k + 1) * HID + jcol];
    c = wmma4(a, b, c);
  }
  const float bcol = bo[jcol];
#pragma unroll
  for (int r = 0; r < 8; ++r) {
    const int M = r + lh * 8;
    const int g = rt * 16 + M;
    out[(size_t)g * HID + jcol] = c[r] + bcol;
  }
}

// ---------------------------------------------------------------------------
extern "C" void kernel_launch(void* const* d_in, const int* in_sizes, int n_in,
                              void* d_out, int out_size, void* d_ws, size_t ws_size,
                              hipStream_t stream) {
  const float* node = (const float*)d_in[0];
  const float* E    = (const float*)d_in[1];
  const unsigned char* mask = (const unsigned char*)d_in[2];
  const float* Wq = (const float*)d_in[3];  const float* bq = (const float*)d_in[4];
  const float* Wk = (const float*)d_in[5];  const float* bk = (const float*)d_in[6];
  const float* Wv = (const float*)d_in[7];  const float* bv = (const float*)d_in[8];
  const float* We = (const float*)d_in[9];  const float* be = (const float*)d_in[10];
  const float* Wo = (const float*)d_in[11]; const float* bo = (const float*)d_in[12];
  float* out = (float*)d_out;

  float* ws   = (float*)d_ws;
  float* Q    = ws;                          // 4*8*512*16   = 262144
  float* K    = ws + 262144;                 // 262144
  float* Vt   = ws + 524288;                 // 262144 ([b][h][d][n])
  float* S    = ws + 786432;                 // 4*8*512*512  = 8388608
  float* attn = ws + 786432 + 8388608;       // 262144

  qkv_proj     <<<dim3(128, 3),     256, 0, stream>>>(node, Wq, bq, Wk, bk, Wv, bv, Q, K, Vt);
  scores_kernel<<<dim3(32, 32, 4),  256, 0, stream>>>(E, mask, We, be, Q, K, S);
  softmax_av   <<<dim3(32, 8, 4),   256, 0, stream>>>(S, Vt, attn);
  out_proj     <<<dim3(128),        256, 0, stream>>>(attn, Wo, bo, out);
}